// HGNN_17884243821250
// MI455X (gfx1250) — compile-verified
//
#include <hip/hip_runtime.h>
#include <hip/hip_bf16.h>

typedef __attribute__((ext_vector_type(2))) float v2f;
typedef __attribute__((ext_vector_type(8))) float v8f;

#define N_NODES  100000
#define N_HEDGES 50000
#define IN_C     128
#define HID_C    128
#define N_CLS    8

// ---------------------------------------------------------------- utilities
__global__ void hg_zero_kernel(float* __restrict__ p, long n) {
  long i = (long)blockIdx.x * blockDim.x + threadIdx.x;
  if (i < n) p[i] = 0.0f;
}

__global__ void hg_deg_kernel(const int* __restrict__ nidx, const int* __restrict__ eidx,
                              float* Dd, float* Bd, int nnz) {
  int i = blockIdx.x * blockDim.x + threadIdx.x;
  if (i < nnz) {
    atomicAdd(&Dd[nidx[i]], 1.0f);
    atomicAdd(&Bd[eidx[i]], 1.0f);
  }
}

__global__ void hg_recip_kernel(float* __restrict__ v, int n) {
  int i = blockIdx.x * blockDim.x + threadIdx.x;
  if (i < n) {
    float x = v[i];
    v[i] = (x > 0.0f) ? (1.0f / x) : 0.0f;
  }
}

// C must be a power of two (128 or 8 here)
__global__ void hg_bias_init_kernel(float* __restrict__ o, const float* __restrict__ b,
                                    long n, int Cmask) {
  long i = (long)blockIdx.x * blockDim.x + threadIdx.x;
  if (i < n) o[i] = b[(int)i & Cmask];
}

__global__ void hg_relu_kernel(float* __restrict__ p, long n) {
  long i = (long)blockIdx.x * blockDim.x + threadIdx.x;
  if (i < n) p[i] = fmaxf(p[i], 0.0f);
}

// zero-pad W2 [128,8] row-major into Wp [128,16]
__global__ void hg_padw_kernel(const float* __restrict__ W2, float* __restrict__ Wp) {
  int i = blockIdx.x * blockDim.x + threadIdx.x;
  if (i < HID_C * 16) {
    int r = i >> 4, c = i & 15;
    Wp[i] = (c < N_CLS) ? W2[r * N_CLS + c] : 0.0f;
  }
}

// ---------------------------------------------------------------- WMMA GEMM
// C[M x ncols] = A[M x 128] * B[128 x ldb], full fp32 via v_wmma_f32_16x16x4_f32.
// Requirements: M % 16 == 0 (true here: 100000), K = 128 fixed.
// B panel (contiguous 128*ldb floats) staged in LDS once per block; the K-loop
// feeds WMMA from ds_load while A rows stream from global as b64 loads.
// Grid mapping: global wave g -> tileN = g % ntilesN, tileM = g / ntilesN,
// so with ntilesN == 8 each block covers one M-tile (A rows reused in L0).
// A-lane layout (16x4 f32): lane L holds A[M=L&15][K = 2*(L>>4) + j], j=0..1
// B-lane layout (4x16 f32): lane L holds B[K = 2*(L>>4) + j][N=L&15]
// C/D layout: VGPR j, lane L -> (row = j + 8*(L>>4), col = L&15)
__global__ void hg_wmma_gemm_f32(const float* __restrict__ A, const float* __restrict__ B,
                                 float* __restrict__ C, int M, int ldb, int ldc,
                                 int ntilesN, int ncols) {
  extern __shared__ float sB[];
  const int nb = 128 * ldb;
  for (int i = threadIdx.x; i < nb; i += blockDim.x) sB[i] = B[i];
  __syncthreads();

  int wave = threadIdx.x >> 5;
  int g = blockIdx.x * (blockDim.x >> 5) + wave;
  int mtiles = M >> 4;
  if (g < mtiles * ntilesN) {
    int tileN = g % ntilesN;
    int tileM = g / ntilesN;

    int lane = threadIdx.x & 31;
    int half = lane >> 4;  // 0 or 1
    int l15  = lane & 15;

    int m = tileM * 16 + l15;
    int n = tileN * 16 + l15;
    const float* __restrict__ arow = A + (size_t)m * 128;

    v8f acc = {};
#pragma unroll 8
    for (int k0 = 0; k0 < 128; k0 += 4) {
      int ka = k0 + 2 * half;
      v2f a;
      a.x = arow[ka];
      a.y = arow[ka + 1];
      v2f b;
      b.x = sB[ka * ldb + n];
      b.y = sB[(ka + 1) * ldb + n];
      acc = __builtin_amdgcn_wmma_f32_16x16x4_f32(
          /*neg_a=*/false, a, /*neg_b=*/false, b,
          /*c_mod=*/(short)0, acc, /*reuse_a=*/false, /*reuse_b=*/false);
    }

    if (n < ncols) {
      float* __restrict__ crow = C + (size_t)(tileM * 16 + half * 8) * ldc + n;
#pragma unroll
      for (int j = 0; j < 8; ++j) crow[(size_t)j * ldc] = acc[j];
    }
  }
}

// ------------------------------------------------------- scatter (C = 128)
// One wave per incidence; each lane carries float4 (128-bit load, 4 atomics).
// out[d*128 + c] += in[s*128 + c] * scale[d]
__global__ void hg_scatter_c128(const float* __restrict__ in, const int* __restrict__ sidx,
                                float* out, const int* __restrict__ didx,
                                const float* __restrict__ scale, int nnz) {
  int w = (blockIdx.x * blockDim.x + threadIdx.x) >> 5;
  if (w >= nnz) return;
  int lane = threadIdx.x & 31;
  int s = sidx[w];
  int d = didx[w];
  float sc = scale[d];
  float4 v = ((const float4*)(in + (size_t)s * 128))[lane];
  float* op = out + (size_t)d * 128 + lane * 4;
  atomicAdd(op + 0, v.x * sc);
  atomicAdd(op + 1, v.y * sc);
  atomicAdd(op + 2, v.z * sc);
  atomicAdd(op + 3, v.w * sc);
}

// --------------------------------------------------------- scatter (C = 8)
__global__ void hg_scatter_c8(const float* __restrict__ in, const int* __restrict__ sidx,
                              float* out, const int* __restrict__ didx,
                              const float* __restrict__ scale, int nnz) {
  long t = (long)blockIdx.x * blockDim.x + threadIdx.x;
  if (t >= (long)nnz * N_CLS) return;
  int w = (int)(t >> 3);
  int c = (int)(t & 7);
  int s = sidx[w];
  int d = didx[w];
  atomicAdd(&out[(size_t)d * N_CLS + c], in[(size_t)s * N_CLS + c] * scale[d]);
}

// ---------------------------------------------------------------- launcher
extern "C" void kernel_launch(void* const* d_in, const int* in_sizes, int n_in,
                              void* d_out, int out_size, void* d_ws, size_t ws_size,
                              hipStream_t stream) {
  (void)n_in; (void)out_size; (void)ws_size;

  const float* x  = (const float*)d_in[0];
  const int*   ei = (const int*)d_in[1];
  const float* W1 = (const float*)d_in[2];
  const float* b1 = (const float*)d_in[3];
  const float* W2 = (const float*)d_in[4];
  const float* b2 = (const float*)d_in[5];
  float* out = (float*)d_out;

  const int nnz = in_sizes[1] / 2;
  const int* node_idx = ei;        // row 0
  const int* edge_idx = ei + nnz;  // row 1

  // workspace layout (floats); xw1 buffer is reused as the layer-1 output h
  float* ws  = (float*)d_ws;
  float* xw1 = ws;                                          // N_NODES*HID_C
  float* ef1 = xw1 + (size_t)N_NODES * HID_C;               // N_HEDGES*HID_C
  float* xw2 = ef1 + (size_t)N_HEDGES * HID_C;              // N_NODES*N_CLS
  float* ef2 = xw2 + (size_t)N_NODES * N_CLS;               // N_HEDGES*N_CLS
  float* Dd  = ef2 + (size_t)N_HEDGES * N_CLS;              // N_NODES
  float* Bd  = Dd + N_NODES;                                // N_HEDGES
  float* Wp  = Bd + N_HEDGES;                               // HID_C*16

  const int TB = 256;

  // zero ef1..Bd (contiguous region)
  long zn = (long)N_HEDGES * HID_C + (long)N_NODES * N_CLS +
            (long)N_HEDGES * N_CLS + N_NODES + N_HEDGES;
  hg_zero_kernel<<<(unsigned)((zn + TB - 1) / TB), TB, 0, stream>>>(ef1, zn);

  // degrees, then reciprocals in place (Dd -> Dinv, Bd -> Binv)
  hg_deg_kernel<<<(nnz + TB - 1) / TB, TB, 0, stream>>>(node_idx, edge_idx, Dd, Bd, nnz);
  hg_recip_kernel<<<(N_NODES + TB - 1) / TB, TB, 0, stream>>>(Dd, N_NODES);
  hg_recip_kernel<<<(N_HEDGES + TB - 1) / TB, TB, 0, stream>>>(Bd, N_HEDGES);

  const int mtiles = N_NODES / 16;  // 6250 (exact)

  // ---- layer 1 ----
  // xw1 = x @ W1  (WMMA f32, B staged in 64KB LDS, one M-tile per block)
  {
    int tiles = mtiles * (HID_C / 16);
    hg_wmma_gemm_f32<<<(tiles + 7) / 8, TB, HID_C * HID_C * sizeof(float), stream>>>(
        x, W1, xw1, N_NODES, HID_C, HID_C, HID_C / 16, HID_C);
  }
  // e_feat1[e] += xw1[v] * Binv[e]
  hg_scatter_c128<<<(nnz + 7) / 8, TB, 0, stream>>>(xw1, node_idx, ef1, edge_idx, Bd, nnz);
  // reuse xw1 as layer-1 output: init with b1, then scatter edge->node
  hg_bias_init_kernel<<<(unsigned)(((long)N_NODES * HID_C + TB - 1) / TB), TB, 0, stream>>>(
      xw1, b1, (long)N_NODES * HID_C, HID_C - 1);
  hg_scatter_c128<<<(nnz + 7) / 8, TB, 0, stream>>>(ef1, edge_idx, xw1, node_idx, Dd, nnz);
  hg_relu_kernel<<<(unsigned)(((long)N_NODES * HID_C + TB - 1) / TB), TB, 0, stream>>>(
      xw1, (long)N_NODES * HID_C);

  // ---- layer 2 ----
  // pad W2 into [128,16], xw2 = h @ Wp (WMMA f32, single N tile, 8KB LDS)
  hg_padw_kernel<<<(HID_C * 16 + TB - 1) / TB, TB, 0, stream>>>(W2, Wp);
  {
    int tiles = mtiles;
    hg_wmma_gemm_f32<<<(tiles + 7) / 8, TB, HID_C * 16 * sizeof(float), stream>>>(
        xw1, Wp, xw2, N_NODES, 16, N_CLS, 1, N_CLS);
  }
  // e_feat2[e] += xw2[v] * Binv[e]
  hg_scatter_c8<<<(unsigned)(((long)nnz * N_CLS + TB - 1) / TB), TB, 0, stream>>>(
      xw2, node_idx, ef2, edge_idx, Bd, nnz);
  // out = b2, then out[v] += e_feat2[e] * Dinv[v]
  hg_bias_init_kernel<<<(unsigned)(((long)N_NODES * N_CLS + TB - 1) / TB), TB, 0, stream>>>(
      out, b2, (long)N_NODES * N_CLS, N_CLS - 1);
  hg_scatter_c8<<<(unsigned)(((long)nnz * N_CLS + TB - 1) / TB), TB, 0, stream>>>(
      ef2, edge_idx, out, node_idx, Dd, nnz);
}